// BipartiteGraphConvolution_60610578481378
// MI455X (gfx1250) — compile-verified
//
#include <hip/hip_runtime.h>
#include <hip/hip_bf16.h>

typedef __attribute__((ext_vector_type(2))) float v2f;
typedef __attribute__((ext_vector_type(8))) float v8f;

#define EMB 16

// ---------------------------------------------------------------------------
// D = X(16x16, row-major, stride ldx) @ W(16x16, row-major) + C
// via 4 chained V_WMMA_F32_16X16X4_F32.
// A layout (32-bit, 16x4): lanes 0-15 hold K=0,1 in v[0],v[1]; lanes 16-31 K=2,3.
// B layout (32-bit, 4x16): lanes 0-15 hold rows K=0,1; lanes 16-31 rows K=2,3.
// ---------------------------------------------------------------------------
__device__ __forceinline__ v8f wmma_tile16(const float* X, int ldx,
                                           const float* __restrict__ W,
                                           v8f c, int lane) {
    const int half = lane >> 4;   // 0: lanes 0-15, 1: lanes 16-31
    const int idx  = lane & 15;   // M for A, N for B
#pragma unroll
    for (int kb = 0; kb < 4; ++kb) {
        const int k0 = 4 * kb + 2 * half;
        v2f a;
        a.x = X[idx * ldx + k0];
        a.y = X[idx * ldx + k0 + 1];
        v2f b;
        b.x = W[(k0    ) * EMB + idx];
        b.y = W[(k0 + 1) * EMB + idx];
        c = __builtin_amdgcn_wmma_f32_16x16x4_f32(false, a, false, b,
                                                  (short)0, c, false, false);
    }
    return c;
}

// ---------------------------------------------------------------------------
// Zero-fill workspace (S accumulator + degree counters).
// ---------------------------------------------------------------------------
__global__ __launch_bounds__(256) void zero_kernel(float* __restrict__ p, int n) {
    int i = blockIdx.x * blockDim.x + threadIdx.x;
    if (i < n) p[i] = 0.0f;
}

// ---------------------------------------------------------------------------
// Y[tile] = X[tile] @ W (+ bias). One wave per 16-row tile; WMMA f32.
// ---------------------------------------------------------------------------
__global__ __launch_bounds__(256) void node_transform_kernel(
    const float* __restrict__ X, const float* __restrict__ W,
    const float* __restrict__ bias, float* __restrict__ Y, int n_tiles) {
    const int gwave = (blockIdx.x * blockDim.x + threadIdx.x) >> 5;  // uniform per wave
    const int lane  = threadIdx.x & 31;
    if (gwave >= n_tiles) return;

    const float* Xt = X + (size_t)gwave * (16 * EMB);
    v8f c = {};
    c = wmma_tile16(Xt, EMB, W, c, lane);

    const int half = lane >> 4, col = lane & 15;
    const float b = bias ? bias[col] : 0.0f;
    float* Yt = Y + (size_t)gwave * (16 * EMB);
#pragma unroll
    for (int r = 0; r < 8; ++r)
        Yt[(r + 8 * half) * EMB + col] = c[r] + b;
}

// ---------------------------------------------------------------------------
// Edge phase: S[r] += relu(Lmsg[l] + Rmsg[r] + ef[e]*W_edge), deg[r] += 1.
// 4 lanes per edge, float4 (b128) gathers: one wave32 covers 8 edges.
// Lmsg/Rmsg/S are L2-resident (6.4 MB each vs 192 MB L2); the f32 lane-atomic
// stream to L2 is the hard floor, so minimize issue overhead around it.
// ---------------------------------------------------------------------------
__global__ __launch_bounds__(256) void edge_kernel(
    const int* __restrict__ li, const int* __restrict__ ri,
    const float* __restrict__ ef, const float* __restrict__ Wedge,
    const float* __restrict__ Lmsg, const float* __restrict__ Rmsg,
    float* __restrict__ S, float* __restrict__ deg, int nE) {
    const int t = blockIdx.x * blockDim.x + threadIdx.x;
    const int e = t >> 2;        // edge id
    const int q = t & 3;         // quarter: elements 4q..4q+3
    if (e >= nE) return;

    // Speculative prefetch of the upcoming edge stream (global_prefetch_b8).
    if (q == 0 && (e & 1023) == 0 && e + 8192 < nE) {
        __builtin_prefetch(li + e + 8192, 0, 1);
        __builtin_prefetch(ri + e + 8192, 0, 1);
        __builtin_prefetch(ef + e + 8192, 0, 1);
    }

    const int l = li[e];
    const int r = ri[e];
    const float w = ef[e];

    const float4 lm = *(const float4*)(Lmsg  + (size_t)l * EMB + 4 * q);
    const float4 rm = *(const float4*)(Rmsg  + (size_t)r * EMB + 4 * q);
    const float4 we = *(const float4*)(Wedge + 4 * q);

    float* Sp = S + (size_t)r * EMB + 4 * q;
    atomicAdd(Sp + 0, fmaxf(lm.x + rm.x + w * we.x, 0.0f));
    atomicAdd(Sp + 1, fmaxf(lm.y + rm.y + w * we.y, 0.0f));
    atomicAdd(Sp + 2, fmaxf(lm.z + rm.z + w * we.z, 0.0f));
    atomicAdd(Sp + 3, fmaxf(lm.w + rm.w + w * we.w, 0.0f));
    if (q == 0) atomicAdd(&deg[r], 1.0f);
}

// ---------------------------------------------------------------------------
// Right-node phase (all WMMA, relayout via wave-private LDS tiles):
//   agg  = S @ W_final + deg*b_final
//   post = relu(agg) @ W_post + b_post
//   t1   = relu(post @ W_out1[0:16] + right @ W_out1[16:32] + b_out1)
//   out  = t1 @ W_out2 + b_out2
// ---------------------------------------------------------------------------
__global__ __launch_bounds__(256) void right_node_kernel(
    const float* __restrict__ S, const float* __restrict__ deg,
    const float* __restrict__ right,
    const float* __restrict__ Wf,  const float* __restrict__ bf,
    const float* __restrict__ Wp,  const float* __restrict__ bp,
    const float* __restrict__ Wo1, const float* __restrict__ bo1,
    const float* __restrict__ Wo2, const float* __restrict__ bo2,
    float* __restrict__ out, int n_tiles) {
    __shared__ float lds[8][16][EMB];   // one 16x16 tile per wave (8 waves/block)

    const int wv    = threadIdx.x >> 5;
    const int lane  = threadIdx.x & 31;
    const int gwave = (blockIdx.x * blockDim.x + threadIdx.x) >> 5;  // uniform per wave
    if (gwave >= n_tiles) return;

    const int half = lane >> 4, col = lane & 15;
    float* myt = &lds[wv][0][0];

    const float* St = S     + (size_t)gwave * (16 * EMB);
    const float* Rt = right + (size_t)gwave * (16 * EMB);

    // agg = S @ W_final + deg ⊗ b_final ; stash relu(agg) in LDS
    v8f c = {};
    c = wmma_tile16(St, EMB, Wf, c, lane);
    const float bfc = bf[col];
#pragma unroll
    for (int r = 0; r < 8; ++r) {
        const float d = deg[(size_t)gwave * 16 + r + 8 * half];
        myt[(r + 8 * half) * EMB + col] = fmaxf(c[r] + d * bfc, 0.0f);
    }

    // post = relu(agg) @ W_post + b_post ; stash in LDS
    v8f c2 = {};
    c2 = wmma_tile16(myt, EMB, Wp, c2, lane);
    const float bpc = bp[col];
#pragma unroll
    for (int r = 0; r < 8; ++r)
        myt[(r + 8 * half) * EMB + col] = c2[r] + bpc;

    // t1 = relu(post @ Wo1_top + right @ Wo1_bot + bo1) ; stash in LDS
    v8f c3 = {};
    c3 = wmma_tile16(myt, EMB, Wo1, c3, lane);            // rows 0..15 of W_out1
    c3 = wmma_tile16(Rt,  EMB, Wo1 + 16 * EMB, c3, lane); // rows 16..31
    const float bo1c = bo1[col];
#pragma unroll
    for (int r = 0; r < 8; ++r)
        myt[(r + 8 * half) * EMB + col] = fmaxf(c3[r] + bo1c, 0.0f);

    // out = t1 @ W_out2 + b_out2
    v8f c4 = {};
    c4 = wmma_tile16(myt, EMB, Wo2, c4, lane);
    const float bo2c = bo2[col];
    float* Ot = out + (size_t)gwave * (16 * EMB);
#pragma unroll
    for (int r = 0; r < 8; ++r)
        Ot[(r + 8 * half) * EMB + col] = c4[r] + bo2c;
}

// ---------------------------------------------------------------------------
extern "C" void kernel_launch(void* const* d_in, const int* in_sizes, int n_in,
                              void* d_out, int out_size, void* d_ws, size_t ws_size,
                              hipStream_t stream) {
    const float* left   = (const float*)d_in[0];
    const int*   eidx   = (const int*)  d_in[1];   // (2, E) int32
    const float* ef     = (const float*)d_in[2];   // (E, 1)
    const float* right  = (const float*)d_in[3];
    const float* W_left = (const float*)d_in[4];
    const float* b_left = (const float*)d_in[5];
    const float* W_edge = (const float*)d_in[6];   // (1, 16)
    const float* W_right= (const float*)d_in[7];
    const float* W_final= (const float*)d_in[8];
    const float* b_final= (const float*)d_in[9];
    const float* W_post = (const float*)d_in[10];
    const float* b_post = (const float*)d_in[11];
    const float* W_out1 = (const float*)d_in[12];  // (32, 16)
    const float* b_out1 = (const float*)d_in[13];
    const float* W_out2 = (const float*)d_in[14];
    const float* b_out2 = (const float*)d_in[15];

    const int nL = in_sizes[0] / EMB;   // 100000
    const int nR = in_sizes[3] / EMB;   // 100000
    const int nE = in_sizes[2];         // 3200000

    const int nLt = (nL + 15) / 16;
    const int nRt = (nR + 15) / 16;

    // Workspace layout (floats): Lmsg | Rmsg | S | deg  (~19.6 MB)
    float* Lmsg = (float*)d_ws;
    float* Rmsg = Lmsg + (size_t)nL * EMB;
    float* Sacc = Rmsg + (size_t)nR * EMB;
    float* degf = Sacc + (size_t)nR * EMB;

    // 1) zero S and deg (contiguous)
    {
        const int n = nR * EMB + nR;
        zero_kernel<<<(n + 255) / 256, 256, 0, stream>>>(Sacc, n);
    }

    // 2) node transforms (WMMA): Lmsg = left@W_left + b_left ; Rmsg = right@W_right
    {
        const int tilesPerBlock = 256 / 32;  // 8 waves/block
        node_transform_kernel<<<(nLt + tilesPerBlock - 1) / tilesPerBlock, 256, 0, stream>>>(
            left, W_left, b_left, Lmsg, nLt);
        node_transform_kernel<<<(nRt + tilesPerBlock - 1) / tilesPerBlock, 256, 0, stream>>>(
            right, W_right, nullptr, Rmsg, nRt);
    }

    // 3) edge scatter: S += relu(msgs), deg += 1  (W_final hoisted out of edge loop)
    {
        const long long threads = (long long)nE * 4;
        edge_kernel<<<(int)((threads + 255) / 256), 256, 0, stream>>>(
            eidx, eidx + nE, ef, W_edge, Lmsg, Rmsg, Sacc, degf, nE);
    }

    // 4) right-node chain (WMMA x5 per tile)
    {
        const int tilesPerBlock = 256 / 32;
        right_node_kernel<<<(nRt + tilesPerBlock - 1) / tilesPerBlock, 256, 0, stream>>>(
            Sacc, degf, right, W_final, b_final, W_post, b_post,
            W_out1, b_out1, W_out2, b_out2, (float*)d_out, nRt);
    }
}